// TopK_32839319945559
// MI455X (gfx1250) — compile-verified
//
#include <hip/hip_runtime.h>
#include <stdint.h>

// Per-row top-K (K=64) + relu + scatter for x[8192][32768] f32.
// One workgroup per row; whole row staged in CDNA5's 320KB LDS via async DMA
// (GLOBAL_LOAD_ASYNC_TO_LDS_B128, non-temporal); 4x8-bit radix select in LDS;
// single HBM read + single non-temporal HBM write (bandwidth roofline ~92us).

#define NFEAT 32768
#define KSEL  64
#define TPB   256
#define CHUNK (NFEAT / TPB)   // 128 elements per thread

typedef float v4f __attribute__((ext_vector_type(4)));   // native vec for NT stores

// Dynamic-LDS byte layout (no static __shared__ anywhere, so dynamic base == 0,
// which the async-to-LDS byte offsets below rely on):
//   [0,      131072)  keys : 32768 x u32 (raw floats first, then monotonic keys)
//   [131072, 132096)  hist : 256 x u32   (radix histogram, reused for tie scan)
//   [132096, 132352)  sel  : 64 x u32    (tie-break winner indices)
//   [132352, 132368)  scal : scalars {bucket, cum}
#define SM_KEYS 0
#define SM_HIST (NFEAT * 4)
#define SM_SEL  (SM_HIST + 256 * 4)
#define SM_SCAL (SM_SEL + KSEL * 4)
#define SMEM_BYTES (SM_SCAL + 16)

__device__ __forceinline__ unsigned f2key(unsigned bits) {
    // order-preserving f32 -> u32 (larger float => larger unsigned key)
    return (bits & 0x80000000u) ? ~bits : (bits | 0x80000000u);
}
__device__ __forceinline__ float key2f(unsigned k) {
    unsigned bits = (k & 0x80000000u) ? (k & 0x7FFFFFFFu) : ~k;
    return __uint_as_float(bits);
}

__device__ __forceinline__ float pick_val(unsigned k, unsigned i, unsigned T,
                                          unsigned R, const unsigned* sel) {
    if (k > T) return fmaxf(key2f(k), 0.0f);
    if (k == T) {  // ties at the threshold: only the R lowest-index ones win
        for (unsigned r = 0; r < R; ++r)
            if (sel[r] == i) return fmaxf(key2f(k), 0.0f);
    }
    return 0.0f;
}

__global__ __launch_bounds__(TPB)
void topk64_scatter_kernel(const float* __restrict__ x, float* __restrict__ out) {
    extern __shared__ unsigned char smem_raw[];
    unsigned* keys = (unsigned*)(smem_raw + SM_KEYS);
    unsigned* hist = (unsigned*)(smem_raw + SM_HIST);
    unsigned* sel  = (unsigned*)(smem_raw + SM_SEL);
    volatile unsigned* scal = (volatile unsigned*)(smem_raw + SM_SCAL);

    const unsigned tid = threadIdx.x;
    const float* xrow = x   + (size_t)blockIdx.x * NFEAT;
    float*       orow = out + (size_t)blockIdx.x * NFEAT;

    // ---- Phase 0: async DMA row -> LDS, non-temporal (read-once stream) ----
    // Per step the workgroup pulls 1024 consecutive floats (4KB burst).
    #pragma unroll
    for (int j = 0; j < CHUNK / 4; ++j) {
        unsigned idx     = (unsigned)j * (TPB * 4) + tid * 4;  // element index
        unsigned lds_off = SM_KEYS + idx * 4;                  // LDS byte addr
        uint64_t gaddr   = (uint64_t)(uintptr_t)(xrow + idx);
        asm volatile("global_load_async_to_lds_b128 %0, %1, off th:TH_LOAD_NT"
                     :: "v"(lds_off), "v"(gaddr) : "memory");
    }
    asm volatile("s_wait_asynccnt 0" ::: "memory");
    __syncthreads();

    // ---- Phase 1: in-place float -> monotonic key (b128 LDS traffic) ----
    #pragma unroll 4
    for (int j = 0; j < CHUNK / 4; ++j) {
        unsigned idx = (unsigned)j * (TPB * 4) + tid * 4;
        uint4 v = *(uint4*)&keys[idx];
        v.x = f2key(v.x); v.y = f2key(v.y); v.z = f2key(v.z); v.w = f2key(v.w);
        *(uint4*)&keys[idx] = v;
    }
    __syncthreads();

    // ---- Phase 2: MSB-first 4x8-bit radix select of the K-th largest key ----
    unsigned prefix = 0, remaining = KSEL, greater = 0;
    for (int pass = 3; pass >= 0; --pass) {
        const unsigned shift   = (unsigned)pass * 8;
        const unsigned mask_hi = (pass == 3) ? 0u : (0xFFFFFFFFu << (shift + 8));
        hist[tid] = 0;              // 256 bins, one per thread
        __syncthreads();
        // b128 LDS reads; atomics fire only for keys matching the prefix
        for (int j = 0; j < CHUNK / 4; ++j) {
            unsigned idx = (unsigned)j * (TPB * 4) + tid * 4;
            uint4 kv = *(const uint4*)&keys[idx];
            if ((kv.x & mask_hi) == prefix) atomicAdd(&hist[(kv.x >> shift) & 0xFFu], 1u);
            if ((kv.y & mask_hi) == prefix) atomicAdd(&hist[(kv.y >> shift) & 0xFFu], 1u);
            if ((kv.z & mask_hi) == prefix) atomicAdd(&hist[(kv.z >> shift) & 0xFFu], 1u);
            if ((kv.w & mask_hi) == prefix) atomicAdd(&hist[(kv.w >> shift) & 0xFFu], 1u);
        }
        __syncthreads();
        if (tid == 0) {             // scan bins from the top down
            unsigned cum = 0; int b = 255;
            for (; b > 0; --b) {
                unsigned h = hist[b];
                if (cum + h >= remaining) break;
                cum += h;
            }
            scal[0] = (unsigned)b;
            scal[1] = cum;
        }
        __syncthreads();
        unsigned b = scal[0], cum = scal[1];
        __syncthreads();
        prefix    |= b << shift;
        remaining -= cum;
        greater   += cum;
    }
    const unsigned T = prefix;          // threshold key (K-th largest)
    const unsigned R = KSEL - greater;  // #keys == T to keep (lowest indices)

    // ---- Phase 3: tie-break — pick R lowest-index keys equal to T ----
    {
        const unsigned base = tid * CHUNK;   // contiguous chunk preserves order
        unsigned cnt = 0;
        for (unsigned j = 0; j < CHUNK; j += 4) {
            uint4 kv = *(const uint4*)&keys[base + j];
            cnt += (kv.x == T) + (kv.y == T) + (kv.z == T) + (kv.w == T);
        }
        hist[tid] = cnt;                     // reuse hist for per-thread counts
        __syncthreads();
        if (tid == 0) {                      // serial exclusive scan, 256 entries
            unsigned acc = 0;
            for (int t = 0; t < TPB; ++t) { unsigned c = hist[t]; hist[t] = acc; acc += c; }
        }
        __syncthreads();
        unsigned r = hist[tid];
        if (r < R) {                         // threads past R slots can't select
            for (unsigned j = 0; j < CHUNK; ++j) {
                if (keys[base + j] == T) {
                    if (r < R) sel[r] = base + j;
                    ++r;
                }
            }
        }
        __syncthreads();
    }

    // ---- Phase 4: single streamed non-temporal write ----
    #pragma unroll 4
    for (int j = 0; j < CHUNK / 4; ++j) {
        unsigned idx = (unsigned)j * (TPB * 4) + tid * 4;
        uint4 kv = *(const uint4*)&keys[idx];
        v4f o;
        o.x = pick_val(kv.x, idx + 0, T, R, sel);
        o.y = pick_val(kv.y, idx + 1, T, R, sel);
        o.z = pick_val(kv.z, idx + 2, T, R, sel);
        o.w = pick_val(kv.w, idx + 3, T, R, sel);
        __builtin_nontemporal_store(o, (v4f*)&orow[idx]);  // th:TH_STORE_NT b128
    }
}

extern "C" void kernel_launch(void* const* d_in, const int* in_sizes, int n_in,
                              void* d_out, int out_size, void* d_ws, size_t ws_size,
                              hipStream_t stream) {
    const float* x = (const float*)d_in[0];
    float* out = (float*)d_out;
    const int rows = in_sizes[0] / NFEAT;   // 8192
    hipLaunchKernelGGL(topk64_scatter_kernel, dim3(rows), dim3(TPB),
                       SMEM_BYTES, stream, x, out);
}